// CrossDomainClassSpecificFrequencyMixStyle_77266461655125
// MI455X (gfx1250) — compile-verified
//
#include <hip/hip_runtime.h>

typedef __attribute__((ext_vector_type(2))) float v2f;
typedef __attribute__((ext_vector_type(4))) float f4;
typedef __attribute__((ext_vector_type(8))) float v8f;

#define B_DIM 64
#define N_DIM 16384
#define C_DIM 64
#define S_CHUNKS 64
#define ROWS_PER_CHUNK 256   // N_DIM / S_CHUNKS
#define F4_PER_B 262144      // N_DIM*C_DIM/4

// ---------------------------------------------------------------------------
// Pass 1: per-(b,c) sum and sum-of-squares via V_WMMA_F32_16X16X4_F32 with an
// all-ones A matrix (D[m][n] += sum_k B[k][n]).  Each block: 8 waves; wave w
// covers channels [(w&3)*16, +16) and rows [half*128, +128) of its 256-row
// chunk, 4 rows per WMMA step, sum and sumsq chains in parallel.
// ---------------------------------------------------------------------------
__global__ __launch_bounds__(256) void stats_kernel(const float* __restrict__ x,
                                                    float* __restrict__ psum,
                                                    float* __restrict__ psq) {
  __shared__ float lds_s[128];
  __shared__ float lds_q[128];
  const int tid   = threadIdx.x;
  const int w     = tid >> 5;        // wave 0..7
  const int lane  = tid & 31;
  const int b     = blockIdx.y;
  const int chunk = blockIdx.x;
  const int chbase = (w & 3) << 4;               // channel group base
  const int n0     = chunk * ROWS_PER_CHUNK + ((w >> 2) * 128);
  const int half   = lane >> 4;                  // B-matrix row pair select
  const int col    = lane & 15;                  // N (channel) within group

  const float* p = x + (size_t)b * (N_DIM * C_DIM)
                     + (size_t)(n0 + 2 * half) * C_DIM + chbase + col;

  v2f a1; a1[0] = 1.0f; a1[1] = 1.0f;            // all-ones A (16x4 f32)
  v8f dsum = {};
  v8f dsq  = {};

  for (int it = 0; it < 32; ++it) {              // 32 * 4 rows = 128 rows
    v2f bv; bv[0] = p[0];         bv[1] = p[C_DIM];
    v2f bq; bq[0] = bv[0]*bv[0];  bq[1] = bv[1]*bv[1];
    dsum = __builtin_amdgcn_wmma_f32_16x16x4_f32(false, a1, false, bv,
                                                 (short)0, dsum, false, false);
    dsq  = __builtin_amdgcn_wmma_f32_16x16x4_f32(false, a1, false, bq,
                                                 (short)0, dsq,  false, false);
    p += 4 * C_DIM;
  }

  // D VGPR0, lanes 0..15: row M=0 of D == column sums for N=lane.
  if (lane < 16) {
    lds_s[w * 16 + lane] = dsum[0];
    lds_q[w * 16 + lane] = dsq[0];
  }
  __syncthreads();

  if (tid < 64) {                                 // one thread per channel
    const int g = tid >> 4, cc = tid & 15;
    const float s = lds_s[g * 16 + cc] + lds_s[(g + 4) * 16 + cc];
    const float q = lds_q[g * 16 + cc] + lds_q[(g + 4) * 16 + cc];
    const size_t o = (((size_t)b * S_CHUNKS) + chunk) * C_DIM + tid;
    psum[o] = s;
    psq[o]  = q;
  }
}

// ---------------------------------------------------------------------------
// Pass 2: reduce the 64 chunk partials -> mu, sig (unbiased var, +eps).
// ---------------------------------------------------------------------------
__global__ __launch_bounds__(64) void combine_kernel(const float* __restrict__ psum,
                                                     const float* __restrict__ psq,
                                                     float* __restrict__ mu,
                                                     float* __restrict__ sig) {
  const int b = blockIdx.x, c = threadIdx.x;
  const float* ps = psum + (size_t)b * (S_CHUNKS * C_DIM) + c;
  const float* pq = psq  + (size_t)b * (S_CHUNKS * C_DIM) + c;
  float s = 0.0f, q = 0.0f;
  for (int k = 0; k < S_CHUNKS; ++k) { s += ps[k * C_DIM]; q += pq[k * C_DIM]; }
  const float inv_n = 1.0f / 16384.0f;
  const float m   = s * inv_n;
  const float var = (q - s * s * inv_n) * (1.0f / 16383.0f);  // ddof=1
  mu[b * C_DIM + c]  = m;
  sig[b * C_DIM + c] = sqrtf(var + 1e-6f);
}

// ---------------------------------------------------------------------------
// Pass 3: partner selection (first-max argmax over valid partners) + fold the
// mix into a per-(b,c) affine: out = x*scale + bias.
// ---------------------------------------------------------------------------
__global__ __launch_bounds__(64) void mixparams_kernel(const float* __restrict__ mu,
                                                       const float* __restrict__ sig,
                                                       const float* __restrict__ alpha_u,
                                                       const float* __restrict__ noise,
                                                       const int* __restrict__ dom,
                                                       const int* __restrict__ cls,
                                                       float* __restrict__ scale,
                                                       float* __restrict__ bias) {
  __shared__ int s_idx;
  const int b = blockIdx.x, c = threadIdx.x;
  if (c == 0) {
    const int cb = cls[b], db = dom[b];
    float best = -1.0f;            // uniform noise is in [0,1)
    int bi = -1;
    for (int j = 0; j < B_DIM; ++j) {
      if (cls[j] == cb && dom[j] != db) {
        const float sc = noise[b * B_DIM + j];
        if (sc > best) { best = sc; bi = j; }   // strict > == first max (jnp.argmax)
      }
    }
    s_idx = (bi < 0) ? b : bi;
  }
  __syncthreads();
  const int j = s_idx;
  const float a  = alpha_u[b] * 0.5f;           // ALPHA_MAX
  const float m1 = mu[b * C_DIM + c],  s1 = sig[b * C_DIM + c];
  const float m2 = mu[j * C_DIM + c],  s2 = sig[j * C_DIM + c];
  const float smix = s1 * a + s2 * (1.0f - a);
  const float mmix = m1 * a + m2 * (1.0f - a);
  const float sc = smix / s1;
  scale[b * C_DIM + c] = sc;
  bias [b * C_DIM + c] = mmix - m1 * sc;
}

// ---------------------------------------------------------------------------
// Pass 4: streaming apply.  float4 (B128) loads, LDS-cached scale/bias, NT
// B128 stores (no reuse of the output).
// ---------------------------------------------------------------------------
__global__ __launch_bounds__(256) void apply_kernel(const float* __restrict__ x,
                                                    const float* __restrict__ scale,
                                                    const float* __restrict__ bias,
                                                    float* __restrict__ out) {
  __shared__ f4 s_sc[16];
  __shared__ f4 s_bi[16];
  const int b   = blockIdx.y;
  const int tid = threadIdx.x;
  if (tid < 16) {
    s_sc[tid] = ((const f4*)(scale + b * C_DIM))[tid];
    s_bi[tid] = ((const f4*)(bias  + b * C_DIM))[tid];
  }
  __syncthreads();
  const f4* xb = (const f4*)x + (size_t)b * F4_PER_B;
  f4*       ob = (f4*)out     + (size_t)b * F4_PER_B;
  const int base = blockIdx.x * 4096 + tid;
#pragma unroll
  for (int k = 0; k < 16; ++k) {
    const int i4 = base + (k << 8);             // coalesced: stride-1 in tid
    const f4 v  = xb[i4];
    const f4 sc = s_sc[i4 & 15];
    const f4 bi = s_bi[i4 & 15];
    const f4 r  = v * sc + bi;
    __builtin_nontemporal_store(r, &ob[i4]);
  }
}

// ---------------------------------------------------------------------------
extern "C" void kernel_launch(void* const* d_in, const int* in_sizes, int n_in,
                              void* d_out, int out_size, void* d_ws, size_t ws_size,
                              hipStream_t stream) {
  const float* x       = (const float*)d_in[0];   // [64,16384,64] f32
  const float* alpha_u = (const float*)d_in[1];   // [64] f32
  const float* noise   = (const float*)d_in[2];   // [64,64] f32
  const int*   dom     = (const int*)d_in[3];     // [64] i32
  const int*   cls     = (const int*)d_in[4];     // [64] i32
  float* out = (float*)d_out;
  float* ws  = (float*)d_ws;

  // workspace layout (floats)
  float* psum  = ws;                // 64*64*64 = 262144
  float* psq   = ws + 262144;       // 262144
  float* mu    = ws + 524288;       // 4096
  float* sig   = ws + 528384;       // 4096
  float* scale = ws + 532480;       // 4096
  float* bias  = ws + 536576;       // 4096   (total ~2.1 MB)

  stats_kernel    <<<dim3(S_CHUNKS, B_DIM), 256, 0, stream>>>(x, psum, psq);
  combine_kernel  <<<B_DIM, C_DIM, 0, stream>>>(psum, psq, mu, sig);
  mixparams_kernel<<<B_DIM, C_DIM, 0, stream>>>(mu, sig, alpha_u, noise, dom, cls, scale, bias);
  apply_kernel    <<<dim3(64, B_DIM), 256, 0, stream>>>(x, scale, bias, out);
}